// Rel_Mot_GNN_4114578669572
// MI455X (gfx1250) — compile-verified
//
#include <hip/hip_runtime.h>
#include <hip/hip_bf16.h>

typedef float v2f __attribute__((ext_vector_type(2)));
typedef float v4f __attribute__((ext_vector_type(4)));
typedef float v8f __attribute__((ext_vector_type(8)));

// ---------------------------------------------------------------------------
// WMMA helper: D = A(16x4 f32) * B(4x16 f32) + C(16x16 f32)
// ---------------------------------------------------------------------------
__device__ __forceinline__ v8f wmma_f32(v2f a, v2f b, v8f c) {
  return __builtin_amdgcn_wmma_f32_16x16x4_f32(false, a, false, b, (short)0, c,
                                               false, false);
}

// Fast SiLU: x * sigmoid(x) with v_exp_f32 + v_rcp_f32 (no IEEE div sequence).
__device__ __forceinline__ float silu_f(float x) {
  float e = __expf(-x);
  return x * __builtin_amdgcn_rcpf(1.0f + e);
}

// ---------------------------------------------------------------------------
// One [16x32] x [32 x (16*NT)] GEMM using V_WMMA_F32_16X16X4_F32.
//   lds_act : 16x32 row-major activation tile in LDS (rows 8..15 are padding)
//   W       : row-major [32][Nw] weights in global memory
//   bias    : [16*NT] bias, broadcast across rows (accumulator init)
// A layout (16x4 f32): lane L holds M=L&15; comp.x = K0+2*(L>>4), comp.y = +1
// B layout (4x16 f32): lane L holds N=L&15 (+tile); same K split as A
// D layout (16x16 f32): VGPR r, lanes 0-15 -> row r, lanes 16-31 -> row r+8
// ---------------------------------------------------------------------------
template <int NT>
__device__ __forceinline__ void mm32(const float* lds_act,
                                     const float* __restrict__ W,
                                     const float* __restrict__ bias, int Nw,
                                     v8f* acc, int lane) {
  const int m = lane & 15;
  const int half = lane >> 4;
  const int ncol = lane & 15;
#pragma unroll
  for (int t = 0; t < NT; ++t) {
    float bv = bias[t * 16 + ncol];
    v8f av;
#pragma unroll
    for (int r = 0; r < 8; ++r) av[r] = bv;
    acc[t] = av;
  }
#pragma unroll
  for (int s = 0; s < 8; ++s) {
    const int k = 4 * s + 2 * half;
    v2f a = *(const v2f*)(lds_act + m * 32 + k);  // ds_load_b64
#pragma unroll
    for (int t = 0; t < NT; ++t) {
      v2f b;
      b.x = W[k * Nw + t * 16 + ncol];
      b.y = W[(k + 1) * Nw + t * 16 + ncol];
      acc[t] = wmma_f32(a, b, acc[t]);
    }
  }
}

// SiLU on both D tiles and write back into the 16x32 LDS activation tile.
__device__ __forceinline__ void silu_store(float* lds_act, const v8f* acc,
                                           int lane) {
  const int half = lane >> 4;
  const int ncol = lane & 15;
#pragma unroll
  for (int t = 0; t < 2; ++t) {
#pragma unroll
    for (int r = 0; r < 8; ++r) {
      float v = silu_f(acc[t][r]);
      lds_act[(r + 8 * half) * 32 + (t * 16 + ncol)] = v;
    }
  }
}

// ---------------------------------------------------------------------------
// Main per-edge-pair kernel: one wave32 per edge pair.
// x  : [N][B=8][F=16] current node features
// xn : [N][8][16] next features (zeroed before launch), scatter via atomics
// Weight pointers are pre-offset to layer l; per-edge strides applied inside.
// ---------------------------------------------------------------------------
__global__ void __launch_bounds__(256) edge_pair_kernel(
    const float* __restrict__ x, float* __restrict__ xn,
    const float* __restrict__ W0, const float* __restrict__ Wh,
    const float* __restrict__ Wout, const float* __restrict__ b0,
    const float* __restrict__ bh, const float* __restrict__ bout,
    const int* __restrict__ src, const int* __restrict__ dst, int P, int H) {
  __shared__ float lds[8][16 * 32];  // 2 KB per wave, 8 waves per block
  const int wave = threadIdx.x >> 5;
  const int lane = threadIdx.x & 31;
  const int p = blockIdx.x * 8 + wave;
  if (p >= P) return;  // wave-uniform: EXEC stays all-ones for live waves

  float* e = lds[wave];
  const int vtx = dst[p];  // eff scatters + onto dst
  const int utx = src[p];  //  and - onto src (dst of the reversed pair)

  // Zero padding rows 8..15 once (real rows overwritten every stage).
#pragma unroll
  for (int i = 0; i < 8; ++i) e[(8 + i) * 32 + lane] = 0.0f;

  // Gather e[b][0:16] = x[dst], e[b][16:32] = x[src]; 128 floats each side.
  {
    const int b = lane >> 2;
    const int f = (lane & 3) * 4;
    v4f xd = *(const v4f*)(x + (size_t)vtx * 128 + lane * 4);
    *(v4f*)(e + b * 32 + f) = xd;
    v4f xs = *(const v4f*)(x + (size_t)utx * 128 + lane * 4);
    *(v4f*)(e + b * 32 + 16 + f) = xs;
  }

  v8f acc[2];
  // Input layer: [16x32] @ [32x32] + b0, SiLU
  mm32<2>(e, W0 + (size_t)p * 1024, b0 + (size_t)p * 32, 32, acc, lane);
  silu_store(e, acc, lane);
  // Hidden layers
  for (int i = 0; i < H; ++i) {
    mm32<2>(e, Wh + ((size_t)i * P + p) * 1024, bh + ((size_t)i * P + p) * 32,
            32, acc, lane);
    silu_store(e, acc, lane);
  }
  // Output layer: [16x32] @ [32x16] + bout (single N tile, no activation)
  mm32<1>(e, Wout + (size_t)p * 512, bout + (size_t)p * 16, 16, acc, lane);

  // Scatter: lanes 0..15 hold real rows b=r, col f=lane in acc[0][r].
  if (lane < 16) {
    float* pv = xn + (size_t)vtx * 128 + lane;
    float* pu = xn + (size_t)utx * 128 + lane;
#pragma unroll
    for (int r = 0; r < 8; ++r) {
      float val = acc[0][r];
      unsafeAtomicAdd(pv + r * 16, val);    // +eff -> dst
      unsafeAtomicAdd(pu + r * 16, -val);   // -eff -> src
    }
  }
}

// ---------------------------------------------------------------------------
// Fixed-edge kernel (shared weights, one edge, one wave).
// ---------------------------------------------------------------------------
__global__ void __launch_bounds__(32) fixed_edge_kernel(
    const float* __restrict__ x, float* __restrict__ xn,
    const float* __restrict__ W0, const float* __restrict__ Wh,
    const float* __restrict__ Wout, const float* __restrict__ b0,
    const float* __restrict__ bh, const float* __restrict__ bout,
    const int* __restrict__ src, const int* __restrict__ dst, int E, int H) {
  __shared__ float e[16 * 32];
  const int lane = threadIdx.x & 31;
  const int vtx = dst[E - 1];
  const int utx = src[E - 1];

#pragma unroll
  for (int i = 0; i < 8; ++i) e[(8 + i) * 32 + lane] = 0.0f;
  {
    const int b = lane >> 2;
    const int f = (lane & 3) * 4;
    v4f xd = *(const v4f*)(x + (size_t)vtx * 128 + lane * 4);
    *(v4f*)(e + b * 32 + f) = xd;
    v4f xs = *(const v4f*)(x + (size_t)utx * 128 + lane * 4);
    *(v4f*)(e + b * 32 + 16 + f) = xs;
  }

  v8f acc[2];
  mm32<2>(e, W0, b0, 32, acc, lane);
  silu_store(e, acc, lane);
  for (int i = 0; i < H; ++i) {
    mm32<2>(e, Wh + (size_t)i * 1024, bh + (size_t)i * 32, 32, acc, lane);
    silu_store(e, acc, lane);
  }
  mm32<1>(e, Wout, bout, 16, acc, lane);

  if (lane < 16) {
    float* pv = xn + (size_t)vtx * 128 + lane;
#pragma unroll
    for (int r = 0; r < 8; ++r) unsafeAtomicAdd(pv + r * 16, acc[0][r]);
  }
}

// ---------------------------------------------------------------------------
// Layout kernels
// ---------------------------------------------------------------------------
__global__ void transpose_in_kernel(const float* __restrict__ h,
                                    float* __restrict__ x, int N) {
  int idx = blockIdx.x * 256 + threadIdx.x;
  if (idx >= N * 128) return;
  int n = idx >> 7;
  int b = (idx >> 4) & 7;
  int f = idx & 15;
  x[idx] = h[((size_t)b * N + n) * 16 + f];  // h[B][N][F] -> x[N][B][F]
}

__global__ void transpose_out_kernel(const float* __restrict__ x,
                                     float* __restrict__ out, int N) {
  int idx = blockIdx.x * 256 + threadIdx.x;
  if (idx >= N * 128) return;
  int f = idx & 15;
  int n = (idx >> 4) % N;
  int b = idx / (N * 16);
  out[idx] = x[(size_t)n * 128 + b * 16 + f];  // x[N][B][F] -> out[B][N][F]
}

__global__ void zero_kernel(float* __restrict__ p, int n) {
  int idx = blockIdx.x * 256 + threadIdx.x;
  if (idx < n) p[idx] = 0.0f;
}

// ---------------------------------------------------------------------------
// Host launcher
// ---------------------------------------------------------------------------
extern "C" void kernel_launch(void* const* d_in, const int* in_sizes, int n_in,
                              void* d_out, int out_size, void* d_ws,
                              size_t ws_size, hipStream_t stream) {
  const float* h = (const float*)d_in[0];
  const float* Wi0 = (const float*)d_in[1];
  const float* Wih = (const float*)d_in[2];
  const float* Wiout = (const float*)d_in[3];
  const float* bi0 = (const float*)d_in[4];
  const float* bih = (const float*)d_in[5];
  const float* biout = (const float*)d_in[6];
  const float* Wf0 = (const float*)d_in[7];
  const float* Wfh = (const float*)d_in[8];
  const float* Wfout = (const float*)d_in[9];
  const float* bf0 = (const float*)d_in[10];
  const float* bfh = (const float*)d_in[11];
  const float* bfout = (const float*)d_in[12];
  const int* src = (const int*)d_in[13];
  const int* dst = (const int*)d_in[14];

  const int B = 8, F = 16, M = 32;
  const int E = in_sizes[13];
  const int P = (E - 1) / 2;
  const int L = in_sizes[1] / (P * 2 * F * M);
  const int H = in_sizes[2] / (L * P * M * M);
  const int N = in_sizes[0] / (B * F);
  const int total = N * B * F;

  float* x0 = (float*)d_ws;
  float* x1 = x0 + (size_t)total;

  transpose_in_kernel<<<(total + 255) / 256, 256, 0, stream>>>(h, x0, N);

  float* cur = x0;
  float* nxt = x1;
  for (int l = 0; l < L; ++l) {
    zero_kernel<<<(total + 255) / 256, 256, 0, stream>>>(nxt, total);
    edge_pair_kernel<<<(P + 7) / 8, 256, 0, stream>>>(
        cur, nxt, Wi0 + (size_t)l * P * 2 * F * M,
        Wih + (size_t)l * H * P * M * M, Wiout + (size_t)l * P * M * F,
        bi0 + (size_t)l * P * M, bih + (size_t)l * H * P * M,
        biout + (size_t)l * P * F, src, dst, P, H);
    fixed_edge_kernel<<<1, 32, 0, stream>>>(
        cur, nxt, Wf0 + (size_t)l * 2 * F * M, Wfh + (size_t)l * H * M * M,
        Wfout + (size_t)l * M * F, bf0 + (size_t)l * M, bfh + (size_t)l * H * M,
        bfout + (size_t)l * F, src, dst, E, H);
    float* t = cur;
    cur = nxt;
    nxt = t;
  }

  transpose_out_kernel<<<(total + 255) / 256, 256, 0, stream>>>(
      cur, (float*)d_out, N);
}